// GCNLayer_81363860455526
// MI455X (gfx1250) — compile-verified
//
#include <hip/hip_runtime.h>

typedef __attribute__((ext_vector_type(2))) float v2f;
typedef __attribute__((ext_vector_type(8))) float v8f;

// ---------------------------------------------------------------------------
// Zero-fill helper (agg lives in d_out, deg lives in d_ws; both need zeroing
// every call since the harness poisons buffers).
// ---------------------------------------------------------------------------
__global__ void gcn_zero_f32(float* __restrict__ p, int n) {
    int i = blockIdx.x * blockDim.x + threadIdx.x;
    if (i < n) p[i] = 0.0f;
}

// ---------------------------------------------------------------------------
// Edge scatter: agg[row] += x[col]; deg[row] += 1.
// 16 threads per edge, each moving a float4 (coalesced 256B per half-wave of a
// given edge). 51.2M f32 atomics total -> L2 atomic units; x/agg are L2
// resident (12.8 MB each vs 192 MB L2).
// ---------------------------------------------------------------------------
__global__ void gcn_scatter(const float* __restrict__ x,
                            const int*   __restrict__ ei,   // [2, E] int32
                            float*       __restrict__ agg,  // [N, 64]
                            float*       __restrict__ deg,  // [N]
                            int E) {
    int tid = blockIdx.x * blockDim.x + threadIdx.x;
    int e   = tid >> 4;
    if (e >= E) return;
    int sub = tid & 15;
    int row = ei[e];
    int col = ei[E + e];

    float4 v = reinterpret_cast<const float4*>(x + (long)col * 64)[sub];
    float* a = agg + (long)row * 64 + sub * 4;
    atomicAdd(a + 0, v.x);
    atomicAdd(a + 1, v.y);
    atomicAdd(a + 2, v.z);
    atomicAdd(a + 3, v.w);
    if (sub == 0) atomicAdd(deg + row, 1.0f);
}

// ---------------------------------------------------------------------------
// Fused normalize + GEMM + bias using V_WMMA_F32_16X16X4_F32.
// One wave computes a 16x64 output block in-place on agg/out:
//   out[m][n] = sum_k (agg[m][k] / (deg[m]+1e-6)) * W[n][k] + bias[n]
// A (16x4 f32): lane&15 = M, lane>>4 picks K pair {k0,k0+1} / {k0+2,k0+3}.
// B (4x16 f32): lane&15 = N, same K-pair split across lane halves.
// C/D (16x16 f32, v8f): M = vgpr_idx + (lane>>4)*8, N = lane&15.
// ---------------------------------------------------------------------------
__global__ void gcn_gemm_wmma(float*       __restrict__ inout,  // agg in, result out
                              const float* __restrict__ deg,    // [N]
                              const float* __restrict__ W,      // [64 out][64 in]
                              const float* __restrict__ bias,   // [64]
                              int N) {
    int wave = (int)((blockIdx.x * blockDim.x + threadIdx.x) >> 5);
    int lane = threadIdx.x & 31;
    int m0   = wave * 16;
    if (m0 >= N) return;                      // wave-uniform exit: EXEC stays all-ones

    int   mr     = lane & 15;
    int   half   = lane >> 4;                 // 0 or 1
    int   m      = m0 + mr;
    int   mc     = (m < N) ? m : (N - 1);     // clamp for ragged tail (N%16==0 here)
    float invdeg = 1.0f / (deg[mc] + 1e-6f);
    const float* arow = inout + (long)mc * 64;

    int nl = lane & 15;
    const float* w0 = W + (long)(0 * 16 + nl) * 64;
    const float* w1 = W + (long)(1 * 16 + nl) * 64;
    const float* w2 = W + (long)(2 * 16 + nl) * 64;
    const float* w3 = W + (long)(3 * 16 + nl) * 64;

    v8f acc0 = {}, acc1 = {}, acc2 = {}, acc3 = {};

#pragma unroll
    for (int k0 = 0; k0 < 64; k0 += 4) {
        int kb = k0 + half * 2;

        v2f a;
        a.x = arow[kb]     * invdeg;
        a.y = arow[kb + 1] * invdeg;

        v2f b0, b1, b2, b3;
        b0.x = w0[kb]; b0.y = w0[kb + 1];
        b1.x = w1[kb]; b1.y = w1[kb + 1];
        b2.x = w2[kb]; b2.y = w2[kb + 1];
        b3.x = w3[kb]; b3.y = w3[kb + 1];

        acc0 = __builtin_amdgcn_wmma_f32_16x16x4_f32(false, a, false, b0, (short)0, acc0, false, false);
        acc1 = __builtin_amdgcn_wmma_f32_16x16x4_f32(false, a, false, b1, (short)0, acc1, false, false);
        acc2 = __builtin_amdgcn_wmma_f32_16x16x4_f32(false, a, false, b2, (short)0, acc2, false, false);
        acc3 = __builtin_amdgcn_wmma_f32_16x16x4_f32(false, a, false, b3, (short)0, acc3, false, false);
    }

    v8f accs[4] = {acc0, acc1, acc2, acc3};
#pragma unroll
    for (int t = 0; t < 4; ++t) {
        int   n  = t * 16 + nl;
        float bv = bias[n];
#pragma unroll
        for (int i = 0; i < 8; ++i) {
            int row = m0 + half * 8 + i;
            if (row < N) inout[(long)row * 64 + n] = accs[t][i] + bv;
        }
    }
}

// ---------------------------------------------------------------------------
// Launch: inputs in setup_inputs() order: x [N*64] f32, edge_index [2*E] i32,
// W [64*64] f32, b [64] f32. Output [N*64] f32. agg is accumulated directly
// in d_out (GEMM rewrites it in place); only deg (N floats) uses d_ws.
// ---------------------------------------------------------------------------
extern "C" void kernel_launch(void* const* d_in, const int* in_sizes, int n_in,
                              void* d_out, int out_size, void* d_ws, size_t ws_size,
                              hipStream_t stream) {
    const float* x  = (const float*)d_in[0];
    const int*   ei = (const int*)d_in[1];
    const float* W  = (const float*)d_in[2];
    const float* b  = (const float*)d_in[3];
    float* out = (float*)d_out;
    float* deg = (float*)d_ws;

    const int N = in_sizes[0] / 64;
    const int E = in_sizes[1] / 2;

    // Zero agg (in d_out) and deg (in d_ws).
    int zn = N * 64;
    gcn_zero_f32<<<(zn + 255) / 256, 256, 0, stream>>>(out, zn);
    gcn_zero_f32<<<(N + 255) / 256, 256, 0, stream>>>(deg, N);

    // Edge scatter: 16 threads per edge.
    long sthreads = (long)E * 16;
    int  sblocks  = (int)((sthreads + 255) / 256);
    gcn_scatter<<<sblocks, 256, 0, stream>>>(x, ei, out, deg, E);

    // Fused normalize + GEMM + bias: one wave per 16 output rows.
    int waves  = (N + 15) / 16;
    int blocks = (waves * 32 + 255) / 256;   // 8 waves per 256-thread block
    gcn_gemm_wmma<<<blocks, 256, 0, stream>>>(out, deg, W, b, N);
}